// HomoFusedDispatchSF_30623116821154
// MI455X (gfx1250) — compile-verified
//
#include <hip/hip_runtime.h>
#include <hip/hip_bf16.h>

// Problem constants (match reference)
#define T_TOK   16384
#define E_EXP   16
#define D_DIM   2048
#define CAP     2048
#define SLOTS_PER_BLOCK 4

typedef __attribute__((ext_vector_type(4))) float v4f;

// ---------------------------------------------------------------------------
// Phase A: per-expert ordered scan of the multi-hot mask.
// One block per expert (16 blocks, 256 threads = 8 waves of 32).
// Produces fully-defined src_tok[E*CAP] / gate[E*CAP] (tail slots -> sentinel)
// and loads[E] = min(column_count, CAP).
// ---------------------------------------------------------------------------
__global__ void moe_dispatch_index_kernel(const int* __restrict__ hot_mask,
                                          const float* __restrict__ score,
                                          int* __restrict__ src_tok,
                                          float* __restrict__ gate,
                                          int* __restrict__ loads_out) {
  const int e    = blockIdx.x;
  const int tid  = threadIdx.x;
  const int lane = tid & 31;      // wave32
  const int wv   = tid >> 5;      // 8 waves per block

  __shared__ int wsum[8];

  int running = 0;  // uniform across block: tokens seen so far in this column

  for (int chunk = 0; chunk < T_TOK / 256; ++chunk) {
    const int t = chunk * 256 + tid;
    const int m = (hot_mask[(size_t)t * E_EXP + e] > 0) ? 1 : 0;

    int prefix, wtotal;
#if __has_builtin(__builtin_amdgcn_ballot_w32)
    {
      unsigned bmask = __builtin_amdgcn_ballot_w32(m != 0);
      prefix = __popc(bmask & ((1u << lane) - 1u));
      wtotal = __popc(bmask);
    }
#else
    {
      __shared__ int mbits[256];
      mbits[tid] = m;
      __syncthreads();
      prefix = 0;
      wtotal = 0;
      const int base = wv * 32;
      for (int i = 0; i < 32; ++i) {
        int v = mbits[base + i];
        wtotal += v;
        if (i < lane) prefix += v;
      }
      __syncthreads();
    }
#endif
    if (lane == 0) wsum[wv] = wtotal;
    __syncthreads();

    int woff = 0, csum = 0;
#pragma unroll
    for (int w = 0; w < 8; ++w) {
      int s = wsum[w];
      if (w < wv) woff += s;
      csum += s;
    }

    const int pos = running + woff + prefix;  // 0-based rank in token order
    if (m && pos < CAP) {
      src_tok[e * CAP + pos] = t;
      gate[e * CAP + pos]    = score[(size_t)t * E_EXP + e];
    }
    running += csum;
    __syncthreads();  // protect wsum for next chunk
  }

  const int count   = running;
  const int clipped = (count < CAP) ? count : CAP;
  if (tid == 0) loads_out[e] = clipped;

  // Tail: empty slots get sentinel src (== T_TOK) and zero gate.
  for (int s = clipped + tid; s < CAP; s += 256) {
    src_tok[e * CAP + s] = T_TOK;
    gate[e * CAP + s]    = 0.0f;
  }
}

// ---------------------------------------------------------------------------
// Phase B: gather rows, scale by gate, stream to output.
// Each block handles SLOTS_PER_BLOCK consecutive slots. 256 threads prefetch
// all source rows (global_prefetch_b8: 4 rows x 64 x 128B lines), then copy
// with b128 loads and NON-TEMPORAL b128 stores so the 256 MB write stream
// does not evict in_flow (128 MB, resident in the 192 MB L2; rows are reused
// up to TOP_K times).
// ---------------------------------------------------------------------------
__global__ void moe_gather_scale_kernel(const float* __restrict__ in_flow,
                                        const int* __restrict__ src_tok,
                                        const float* __restrict__ gate,
                                        float* __restrict__ out) {
  const int slot0 = blockIdx.x * SLOTS_PER_BLOCK;
  const int tid   = threadIdx.x;

  // Prefetch: thread tid covers row (tid>>6), cacheline (tid&63) of that row.
  {
    const int r    = tid >> 6;                   // 0..3
    const int line = tid & 63;                   // 0..63  (64 x 128B = 8KB row)
    const int psrc = src_tok[slot0 + r];
    if (psrc < T_TOK) {
      __builtin_prefetch(in_flow + (size_t)psrc * D_DIM + (size_t)line * 32, 0, 3);
    }
  }

#pragma unroll
  for (int s = 0; s < SLOTS_PER_BLOCK; ++s) {
    const int   slot = slot0 + s;
    const int   src  = src_tok[slot];
    const float g    = gate[slot];
    v4f* orow = (v4f*)(out + (size_t)slot * D_DIM);

    if (src < T_TOK) {
      const v4f* irow = (const v4f*)(in_flow + (size_t)src * D_DIM);
#pragma unroll
      for (int i = 0; i < D_DIM / 4 / 256; ++i) {  // 2 iterations
        v4f v = irow[i * 256 + tid];
        v *= g;
        __builtin_nontemporal_store(v, &orow[i * 256 + tid]);
      }
    } else {
      v4f z = (v4f)0.0f;
#pragma unroll
      for (int i = 0; i < D_DIM / 4 / 256; ++i) {
        __builtin_nontemporal_store(z, &orow[i * 256 + tid]);
      }
    }
  }
}

// ---------------------------------------------------------------------------
extern "C" void kernel_launch(void* const* d_in, const int* in_sizes, int n_in,
                              void* d_out, int out_size, void* d_ws, size_t ws_size,
                              hipStream_t stream) {
  const float* in_flow  = (const float*)d_in[0];  // [T, D] fp32
  const int*   hot_mask = (const int*)d_in[1];    // [T, E] int32
  const float* score    = (const float*)d_in[2];  // [T, E] fp32

  float* out_rows  = (float*)d_out;                                   // [E*CAP, D]
  int*   loads_out = (int*)d_out + (size_t)E_EXP * CAP * D_DIM;       // [E] int32 tail

  // Workspace: src_tok[E*CAP] ints, then gate[E*CAP] floats (256 KB total).
  int*   src_tok = (int*)d_ws;
  float* gatebuf = (float*)((char*)d_ws + (size_t)E_EXP * CAP * sizeof(int));

  // Phase A: one block per expert.
  moe_dispatch_index_kernel<<<E_EXP, 256, 0, stream>>>(hot_mask, score, src_tok,
                                                       gatebuf, loads_out);

  // Phase B: stream the [E*CAP, D] output.
  const int nblocks = (E_EXP * CAP) / SLOTS_PER_BLOCK;  // 8192
  moe_gather_scale_kernel<<<nblocks, 256, 0, stream>>>(in_flow, src_tok, gatebuf,
                                                       out_rows);
}